// GlobalAgg_36696200577675
// MI455X (gfx1250) — compile-verified
//
#include <hip/hip_runtime.h>
#include <math.h>

// ---------------------------------------------------------------------------
// Shapes (fixed by the reference): B=4, C=512, NH=8, hd=64, D=H=W=32, cr=32
// ---------------------------------------------------------------------------
#define Bv   4
#define Cv   512
#define Lv   32          // D == H == W == sequence length
#define NHv  8
#define HDv  64
#define ROWS (Bv * Lv)   // 128 rows for all GEMMs

typedef __attribute__((ext_vector_type(2))) float v2f;
typedef __attribute__((ext_vector_type(4))) float v4f;
typedef __attribute__((ext_vector_type(8))) float v8f;

// ---------------------------------------------------------------------------
// Kernel 1: triple axis-pooling of f (B,C,D,H,W).  One block per (b,c) slice.
// 256 threads: 32 d-groups x 8 threads; each thread owns 4 h-rows (128 elems).
// ---------------------------------------------------------------------------
__global__ void pool_kernel(const float* __restrict__ f,
                            float* __restrict__ dp,
                            float* __restrict__ hp,
                            float* __restrict__ wp) {
    const int bc = blockIdx.x;                 // b*512 + c
    const float* __restrict__ src = f + (size_t)bc * 32768;
    __shared__ float sd[32], sh[32], sw[32];
    const int t = threadIdx.x;
    if (t < 32) { sd[t] = 0.f; sh[t] = 0.f; sw[t] = 0.f; }
    __syncthreads();

    const int g = t >> 3;                      // d index 0..31
    const int j = t & 7;                       // sub-lane 0..7
    float wacc[32];
#pragma unroll
    for (int i = 0; i < 32; ++i) wacc[i] = 0.f;
    float dsum = 0.f;

#pragma unroll
    for (int rr = 0; rr < 4; ++rr) {
        const int h = j + rr * 8;
        const v4f* __restrict__ row = (const v4f*)(src + g * 1024 + h * 32);
        float rsum = 0.f;
#pragma unroll
        for (int q = 0; q < 8; ++q) {
            v4f v = row[q];
            rsum += v.x + v.y + v.z + v.w;
            wacc[q * 4 + 0] += v.x;
            wacc[q * 4 + 1] += v.y;
            wacc[q * 4 + 2] += v.z;
            wacc[q * 4 + 3] += v.w;
        }
        dsum += rsum;
        atomicAdd(&sh[h], rsum);
    }
    atomicAdd(&sd[g], dsum);
#pragma unroll
    for (int wq = 0; wq < 32; ++wq) atomicAdd(&sw[wq], wacc[wq]);
    __syncthreads();

    if (t < 32) {
        const float inv = 1.0f / 1024.0f;
        dp[bc * 32 + t] = sd[t] * inv;
        hp[bc * 32 + t] = sh[t] * inv;
        wp[bc * 32 + t] = sw[t] * inv;
    }
}

// ---------------------------------------------------------------------------
// Kernel 2: fusion GEMM + LayerNorm + leaky + three sigmoid gates.
// One block per batch, 256 threads.  fused is (32 x 96) in LDS.
// Gate outputs xw/yw/zw stored as rows (b*32+l) x 512 (GEMM row layout).
// ---------------------------------------------------------------------------
__global__ void gate_kernel(const float* __restrict__ dp,
                            const float* __restrict__ hp,
                            const float* __restrict__ wp,
                            const float* __restrict__ fusion_w,  // (32,512)
                            const float* __restrict__ fusion_b,  // (32)
                            const float* __restrict__ depth_w,   // (512,32)
                            const float* __restrict__ depth_b,   // (512)
                            const float* __restrict__ height_w,
                            const float* __restrict__ height_b,
                            const float* __restrict__ width_w,
                            const float* __restrict__ width_b,
                            float* __restrict__ xw,
                            float* __restrict__ yw,
                            float* __restrict__ zw) {
    const int b = blockIdx.x;
    const int t = threadIdx.x;
    __shared__ float fused[32 * 96];

    // fused[o][l] = sum_c fusion_w[o][c] * seq[b][c][l] + fusion_b[o]
    for (int idx = t; idx < 32 * 96; idx += 256) {
        const int o = idx / 96;
        const int l = idx % 96;
        const float* __restrict__ sp = (l < 32) ? dp : (l < 64) ? hp : wp;
        const int ll = (l < 32) ? l : (l < 64) ? (l - 32) : (l - 64);
        const float* __restrict__ wrow = fusion_w + o * Cv;
        float acc = fusion_b[o];
        for (int c = 0; c < Cv; ++c)
            acc += wrow[c] * sp[((size_t)b * Cv + c) * 32 + ll];
        fused[idx] = acc;
    }
    __syncthreads();

    // LayerNorm over l (96) + leaky relu, one row per thread (t<32)
    if (t < 32) {
        float* row = fused + t * 96;
        float mu = 0.f;
        for (int l = 0; l < 96; ++l) mu += row[l];
        mu *= (1.0f / 96.0f);
        float var = 0.f;
        for (int l = 0; l < 96; ++l) { float d = row[l] - mu; var += d * d; }
        var *= (1.0f / 96.0f);
        const float rstd = rsqrtf(var + 1e-5f);
        for (int l = 0; l < 96; ++l) {
            float v = (row[l] - mu) * rstd;
            row[l] = (v >= 0.f) ? v : 0.01f * v;
        }
    }
    __syncthreads();

    // gates: out[b][l][o] = sigmoid( sum_k W[o][k] * fused[k][gate*32 + l] + bias[o] )
    for (int idx = t; idx < 3 * Lv * Cv; idx += 256) {
        const int gate = idx / (Lv * Cv);
        const int rem  = idx % (Lv * Cv);
        const int l = rem / Cv;
        const int o = rem % Cv;
        const float* __restrict__ W  = (gate == 0) ? depth_w  : (gate == 1) ? height_w : width_w;
        const float* __restrict__ bb = (gate == 0) ? depth_b  : (gate == 1) ? height_b : width_b;
        float* __restrict__ dst      = (gate == 0) ? xw : (gate == 1) ? yw : zw;
        float acc = bb[o];
        const float* __restrict__ wrow = W + o * 32;
#pragma unroll
        for (int k = 0; k < 32; ++k)
            acc += wrow[k] * fused[k * 96 + gate * 32 + l];
        dst[((size_t)b * Lv + l) * Cv + o] = 1.0f / (1.0f + __expf(-acc));
    }
}

// ---------------------------------------------------------------------------
// Kernel 3: WMMA-f32 GEMM   OUT[r][o] = sum_c A[r][c] * W[o][c] + bias[o]
// A rows: 128 (=B*L), K = 512, N columns = 1536 (QKV) or 512 (out-proj).
// Columns [0,512) read from qsrc, columns [512,N) from kvsrc (QKV split);
// pass kvsrc == qsrc for a plain GEMM.  One wave per 16x16 tile, K-step 4
// via V_WMMA_F32_16X16X4_F32 (full f32 precision — reference is f32 and
// this section is latency-, not throughput-, bound).
// Grid: N/16 blocks x 256 threads (8 waves -> 8 adjacent column-tiles of one row-tile).
// ---------------------------------------------------------------------------
__global__ void proj_kernel(const float* __restrict__ qsrc,
                            const float* __restrict__ kvsrc,
                            const float* __restrict__ W,      // (N,512) row-major
                            const float* __restrict__ bias,   // (N)
                            float* __restrict__ out,          // (128,N)
                            int N) {
    const int lane = threadIdx.x & 31;
    const int wave = threadIdx.x >> 5;
    const int colTiles = N >> 4;
    const int tile = blockIdx.x * 8 + wave;        // 0 .. 8*colTiles-1
    const int rowT = tile / colTiles;              // 0..7
    const int colT = tile % colTiles;
    const int row0 = rowT * 16;
    const int col0 = colT * 16;

    const float* __restrict__ src = (col0 < Cv) ? qsrc : kvsrc;  // uniform per wave
    const float* __restrict__ A  = src + (size_t)row0 * Cv;      // ld = 512
    const float* __restrict__ Wt = W + (size_t)col0 * Cv;        // ld = 512

    const int half = lane >> 4;        // 0: K pair {0,1}, 1: K pair {2,3}
    const int lr   = lane & 15;        // row (A) / col (B) within tile

    v8f acc = {0.f, 0.f, 0.f, 0.f, 0.f, 0.f, 0.f, 0.f};

    for (int k = 0; k < Cv; k += 4) {
        v2f a, bfr;
        const int kk = k + half * 2;
        a.x   = A[lr * Cv + kk + 0];
        a.y   = A[lr * Cv + kk + 1];
        bfr.x = Wt[lr * Cv + kk + 0];
        bfr.y = Wt[lr * Cv + kk + 1];
        acc = __builtin_amdgcn_wmma_f32_16x16x4_f32(
                  /*neg_a=*/false, a, /*neg_b=*/false, bfr,
                  /*c_mod=*/(short)0, acc, /*reuse_a=*/false, /*reuse_b=*/false);
    }

    const float bv = bias[col0 + lr];
#pragma unroll
    for (int r = 0; r < 8; ++r) {
        const int row = row0 + r + half * 8;       // D layout: VGPR r = M r / r+8
        out[(size_t)row * N + col0 + lr] = acc[r] + bv;
    }
}

// ---------------------------------------------------------------------------
// Kernel 4: attention core per (b, head).  qkv is (128 x 1536):
// q at col 0, k at 512, v at 1024.  ctx is (128 x 512).
// ---------------------------------------------------------------------------
__global__ void attn_kernel(const float* __restrict__ qkv,
                            float* __restrict__ ctx) {
    const int b = blockIdx.x >> 3;
    const int h = blockIdx.x & 7;
    const int t = threadIdx.x;
    __shared__ float qs[Lv * HDv], ks[Lv * HDv], vs[Lv * HDv];
    __shared__ float sc[Lv * Lv];

    for (int idx = t; idx < Lv * HDv; idx += 256) {
        const int l = idx >> 6, d = idx & 63;
        const size_t base = ((size_t)b * Lv + l) * 1536 + h * HDv + d;
        qs[idx] = qkv[base];
        ks[idx] = qkv[base + 512];
        vs[idx] = qkv[base + 1024];
    }
    __syncthreads();

    for (int idx = t; idx < Lv * Lv; idx += 256) {
        const int q = idx >> 5, k = idx & 31;
        float a = 0.f;
#pragma unroll
        for (int d = 0; d < HDv; ++d) a += qs[q * HDv + d] * ks[k * HDv + d];
        sc[idx] = a * 0.125f;                      // 1/sqrt(64)
    }
    __syncthreads();

    if (t < Lv) {                                  // row-wise softmax
        float* row = sc + t * Lv;
        float m = row[0];
        for (int k = 1; k < Lv; ++k) m = fmaxf(m, row[k]);
        float s = 0.f;
        for (int k = 0; k < Lv; ++k) { float e = __expf(row[k] - m); row[k] = e; s += e; }
        const float inv = 1.0f / s;
        for (int k = 0; k < Lv; ++k) row[k] *= inv;
    }
    __syncthreads();

    for (int idx = t; idx < Lv * HDv; idx += 256) {
        const int l = idx >> 6, d = idx & 63;
        float a = 0.f;
#pragma unroll
        for (int k = 0; k < Lv; ++k) a += sc[l * Lv + k] * vs[k * HDv + d];
        ctx[((size_t)b * Lv + l) * Cv + h * HDv + d] = a;
    }
}

// ---------------------------------------------------------------------------
// Kernel 5: out = f * ax * ay * az  (broadcast over d/h/w respectively).
// axf/ayf/azf are (b*32+l, c); factor for (b,c,axis=l) = buf[(b*32+l)*512+c].
// One block per (b,c) slice; streaming non-temporal float4 traffic.
// ---------------------------------------------------------------------------
__global__ void final_kernel(const float* __restrict__ f,
                             const float* __restrict__ axf,
                             const float* __restrict__ ayf,
                             const float* __restrict__ azf,
                             float* __restrict__ out) {
    const int bc = blockIdx.x;
    const int b = bc >> 9;
    const int c = bc & 511;
    const int t = threadIdx.x;
    __shared__ float fd[32], fh[32], fw[32];
    if (t < 32) {
        fd[t] = axf[((size_t)b * Lv + t) * Cv + c];
        fh[t] = ayf[((size_t)b * Lv + t) * Cv + c];
        fw[t] = azf[((size_t)b * Lv + t) * Cv + c];
    }
    __syncthreads();

    const v4f* __restrict__ src = (const v4f*)(f + (size_t)bc * 32768);
    v4f* __restrict__ dst = (v4f*)(out + (size_t)bc * 32768);
    for (int i = t; i < 8192; i += 256) {
        const int e = i << 2;
        const int d = e >> 10, h = (e >> 5) & 31, w0 = e & 31;
        v4f v = __builtin_nontemporal_load(&src[i]);
        const float s = fd[d] * fh[h];
        v.x *= s * fw[w0 + 0];
        v.y *= s * fw[w0 + 1];
        v.z *= s * fw[w0 + 2];
        v.w *= s * fw[w0 + 3];
        __builtin_nontemporal_store(v, &dst[i]);
    }
}

// ---------------------------------------------------------------------------
// Host orchestration
// ---------------------------------------------------------------------------
static void run_mha(const float* qsrc, const float* kvsrc,
                    const float* Wi, const float* bi,
                    const float* Wo, const float* bo,
                    float* qkv, float* ctx, float* dst, hipStream_t stream) {
    proj_kernel<<<dim3(1536 / 16), dim3(256), 0, stream>>>(qsrc, kvsrc, Wi, bi, qkv, 1536);
    attn_kernel<<<dim3(Bv * NHv), dim3(256), 0, stream>>>(qkv, ctx);
    proj_kernel<<<dim3(512 / 16), dim3(256), 0, stream>>>(ctx, ctx, Wo, bo, dst, 512);
}

extern "C" void kernel_launch(void* const* d_in, const int* in_sizes, int n_in,
                              void* d_out, int out_size, void* d_ws, size_t ws_size,
                              hipStream_t stream) {
    const float* f        = (const float*)d_in[0];
    const float* fusion_w = (const float*)d_in[1];
    const float* fusion_b = (const float*)d_in[2];
    const float* depth_w  = (const float*)d_in[3];
    const float* depth_b  = (const float*)d_in[4];
    const float* height_w = (const float*)d_in[5];
    const float* height_b = (const float*)d_in[6];
    const float* width_w  = (const float*)d_in[7];
    const float* width_b  = (const float*)d_in[8];
    const float* wi_x = (const float*)d_in[9];
    const float* bi_x = (const float*)d_in[10];
    const float* wo_x = (const float*)d_in[11];
    const float* bo_x = (const float*)d_in[12];
    const float* wi_y = (const float*)d_in[13];
    const float* bi_y = (const float*)d_in[14];
    const float* wo_y = (const float*)d_in[15];
    const float* bo_y = (const float*)d_in[16];
    const float* wi_z = (const float*)d_in[17];
    const float* bi_z = (const float*)d_in[18];
    const float* wo_z = (const float*)d_in[19];
    const float* bo_z = (const float*)d_in[20];
    float* out = (float*)d_out;

    // Workspace layout (floats)
    float* ws  = (float*)d_ws;
    float* dp  = ws +      0;   // 4*512*32
    float* hp  = ws +  65536;
    float* wp  = ws + 131072;
    float* xw  = ws + 196608;   // (b*32+l, 512)
    float* yw  = ws + 262144;
    float* zw  = ws + 327680;
    float* qkv = ws + 393216;   // 128 x 1536
    float* ctx = ws + 589824;   // 128 x 512
    float* tmp = ws + 655360;   // 128 x 512
    float* axf = ws + 720896;   // 128 x 512
    float* ayf = ws + 786432;
    float* azf = ws + 851968;

    // 1) pooled stats
    pool_kernel<<<dim3(Bv * Cv), dim3(256), 0, stream>>>(f, dp, hp, wp);

    // 2) fusion + LN + gates
    gate_kernel<<<dim3(Bv), dim3(256), 0, stream>>>(dp, hp, wp,
        fusion_w, fusion_b, depth_w, depth_b, height_w, height_b,
        width_w, width_b, xw, yw, zw);

    // 3) six MHA calls (WMMA GEMMs + attention core)
    run_mha(xw,  yw, wi_x, bi_x, wo_x, bo_x, qkv, ctx, tmp, stream);
    run_mha(tmp, zw, wi_x, bi_x, wo_x, bo_x, qkv, ctx, axf, stream);

    run_mha(yw,  xw, wi_y, bi_y, wo_y, bo_y, qkv, ctx, tmp, stream);
    run_mha(tmp, zw, wi_y, bi_y, wo_y, bo_y, qkv, ctx, ayf, stream);

    run_mha(zw,  xw, wi_z, bi_z, wo_z, bo_z, qkv, ctx, tmp, stream);
    run_mha(tmp, yw, wi_z, bi_z, wo_z, bo_z, qkv, ctx, azf, stream);

    // 4) broadcast multiply (streaming, non-temporal)
    final_kernel<<<dim3(Bv * Cv), dim3(256), 0, stream>>>(f, axf, ayf, azf, out);
}